// PopulationLayer_37495064494224
// MI455X (gfx1250) — compile-verified
//
#include <hip/hip_runtime.h>

typedef float v2f __attribute__((ext_vector_type(2)));
typedef float v8f __attribute__((ext_vector_type(8)));

#define B_DIM 32
#define P_DIM 8192
#define NCHUNK 16
#define KCHUNK (P_DIM / NCHUNK)            // 512 rows of W per K-chunk
#define WAVES_PER_BLOCK 8
#define COLS_PER_BLOCK (WAVES_PER_BLOCK * 16)  // 128 output columns per block

#define DT_C 1.0f
#define TAU_MEM_C 20.0f
#define THETA0_C 1.0f
#define TAU_THETA_C 100.0f
#define THETA_PLUS_C 0.05f
#define HOMEO_ETA_C 0.01f
#define TARGET_RATE_C 0.05f

// ---------------------------------------------------------------------------
// Kernel 1: partial GEMV  y_c[j] = sum_{k in chunk c} pop[k] * W[k][j]
// via V_WMMA_F32_16X16X4_F32: B carries the streamed 4x16 W tile (256 useful
// bytes per wmma), A carries pop[k..k+3] replicated across all 16 M-rows, so
// every row of the f32 accumulator holds the same 16 column partials.
// Two independent accumulators break the XDL C->D dependency chain; unroll 8
// keeps ~16 dword loads in flight per wave for HBM latency coverage.
// W is streamed once (256 MB) with non-temporal loads (W > 192 MB L2).
// ---------------------------------------------------------------------------
__global__ __launch_bounds__(256) void pop_gemv_wmma(
    const float* __restrict__ W, const float* __restrict__ pop,
    float* __restrict__ partials)
{
    __shared__ float pop_lds[KCHUNK];

    const int tid = threadIdx.x;
    const int kc  = blockIdx.y * KCHUNK;

    // stage this K-chunk of the activity vector in LDS (2 KB)
    for (int i = tid; i < KCHUNK; i += 256) pop_lds[i] = pop[kc + i];
    __syncthreads();

    const int lane = tid & 31;
    const int wave = tid >> 5;
    const int half = lane >> 4;                 // K msb per CDNA5 f32 A/B layout
    const int n    = blockIdx.x * COLS_PER_BLOCK + wave * 16 + (lane & 15);

    v8f acc0 = {};
    v8f acc1 = {};

    #pragma unroll 8
    for (int k = 0; k < KCHUNK; k += 4) {
        const int r = k + half * 2;             // this half-wave's 2 K rows
        v2f b, a;
        // B[k_local][n] : two rows of the same column, 16-lane coalesced
        b.x = __builtin_nontemporal_load(&W[(size_t)(kc + r)     * P_DIM + n]);
        b.y = __builtin_nontemporal_load(&W[(size_t)(kc + r + 1) * P_DIM + n]);
        // A[m][k_local] = pop[kc + k + k_local]  (replicated over M)
        a.x = pop_lds[r];
        a.y = pop_lds[r + 1];
        // D = A*B + C  (8 args: neg_a, A, neg_b, B, c_mod, C, reuse_a, reuse_b)
        if ((k >> 2) & 1) {
            acc1 = __builtin_amdgcn_wmma_f32_16x16x4_f32(
                false, a, false, b, (short)0, acc1, false, false);
        } else {
            acc0 = __builtin_amdgcn_wmma_f32_16x16x4_f32(
                false, a, false, b, (short)0, acc0, false, false);
        }
    }

    // every row of D is identical; lanes 0-15 VGPR0 hold (M=0, N=lane)
    if (lane < 16)
        partials[(size_t)blockIdx.y * P_DIM + n] = acc0[0] + acc1[0];
}

// ---------------------------------------------------------------------------
// Kernel 2: deterministic partial reduction + diagonal fix + fused LIF update
// One thread per neuron column j; loops the 32 batch rows (coalesced).
// ---------------------------------------------------------------------------
__global__ __launch_bounds__(256) void lif_fused(
    const float* __restrict__ ext,  const float* __restrict__ v_in,
    const float* __restrict__ th_in, const float* __restrict__ W,
    const float* __restrict__ pop,  const float* __restrict__ hb_in,
    const float* __restrict__ partials, float* __restrict__ out)
{
    const int j = blockIdx.x * blockDim.x + threadIdx.x;
    if (j >= P_DIM) return;

    float y = 0.0f;
    #pragma unroll
    for (int c = 0; c < NCHUNK; ++c)
        y += partials[(size_t)c * P_DIM + j];
    // mask the diagonal: remove pop[j] * W[j][j], then negate (inhibition)
    y -= pop[j] * W[(size_t)j * P_DIM + j];
    const float lateral = -y;
    const float hb = hb_in[j];

    float* __restrict__ spikes_o = out;
    float* __restrict__ v_o      = out + 1 * B_DIM * P_DIM;
    float* __restrict__ th_o     = out + 2 * B_DIM * P_DIM;
    float* __restrict__ ti_o     = out + 3 * B_DIM * P_DIM;
    float* __restrict__ ca_o     = out + 4 * B_DIM * P_DIM;
    float* __restrict__ hb_o     = ca_o + P_DIM;

    float ssum = 0.0f;
    #pragma unroll 4
    for (int b = 0; b < B_DIM; ++b) {
        const size_t idx = (size_t)b * P_DIM + j;
        const float ti   = ext[idx] + lateral + hb;
        const float vold = v_in[idx];
        const float vn   = vold + DT_C * (-vold / TAU_MEM_C + ti);
        const float th   = th_in[idx];
        const float s    = (vn >= th) ? 1.0f : 0.0f;
        spikes_o[idx] = s;
        v_o[idx]      = vn * (1.0f - s);
        th_o[idx]     = th + DT_C * (-(th - THETA0_C) / TAU_THETA_C) + s * THETA_PLUS_C;
        ti_o[idx]     = ti;
        ssum += s;
    }
    const float ca = ssum * (1.0f / (float)B_DIM);
    ca_o[j] = ca;
    hb_o[j] = hb + HOMEO_ETA_C * (TARGET_RATE_C - ca);
}

extern "C" void kernel_launch(void* const* d_in, const int* in_sizes, int n_in,
                              void* d_out, int out_size, void* d_ws, size_t ws_size,
                              hipStream_t stream) {
    const float* ext = (const float*)d_in[0];   // external_input [B,P]
    const float* v   = (const float*)d_in[1];   // v              [B,P]
    const float* th  = (const float*)d_in[2];   // theta          [B,P]
    const float* W   = (const float*)d_in[3];   // lateral_weights[P,P]
    const float* pop = (const float*)d_in[4];   // population_activity [P]
    const float* hb  = (const float*)d_in[5];   // homeo_bias     [P]
    float* out = (float*)d_out;
    float* partials = (float*)d_ws;             // NCHUNK * P floats (512 KB)

    dim3 g1(P_DIM / COLS_PER_BLOCK, NCHUNK);    // 64 x 16 = 1024 blocks
    pop_gemv_wmma<<<g1, 256, 0, stream>>>(W, pop, partials);

    lif_fused<<<P_DIM / 256, 256, 0, stream>>>(ext, v, th, W, pop, hb,
                                               partials, out);
}